// GATNet_5884105195910
// MI455X (gfx1250) — compile-verified
//
#include <hip/hip_runtime.h>
#include <math.h>

typedef __attribute__((ext_vector_type(2))) float v2f;
typedef __attribute__((ext_vector_type(8))) float v8f;

#define DEVFN static __device__ __forceinline__

// Order-preserving float -> uint encoding so we can use native u32 atomic max
DEVFN unsigned encodeOrdered(float f) {
  unsigned u = __float_as_uint(f);
  return (u >> 31) ? ~u : (u | 0x80000000u);
}
DEVFN float decodeOrdered(unsigned u) {
  return __uint_as_float((u >> 31) ? (u & 0x7FFFFFFFu) : ~u);
}

// ---------------------------------------------------------------------------
// Utility: zero fill (grid-stride)
// ---------------------------------------------------------------------------
__global__ void zero_kernel(float* __restrict__ p, long n) {
  long i = (long)blockIdx.x * blockDim.x + threadIdx.x;
  long stride = (long)gridDim.x * blockDim.x;
  for (; i < n; i += stride) p[i] = 0.0f;
}

// ---------------------------------------------------------------------------
// mean(edge_attr, axis=0): LDS partial reduce (ds_add_f32) + 3 global atomics
// ---------------------------------------------------------------------------
__global__ void mean_ea_kernel(const float* __restrict__ ea, float* __restrict__ mean_out, int E) {
  __shared__ float s[3];
  if (threadIdx.x < 3) s[threadIdx.x] = 0.0f;
  __syncthreads();
  long n = (long)E * 3;
  float inv = 1.0f / (float)E;
  long i = (long)blockIdx.x * blockDim.x + threadIdx.x;
  long stride = (long)gridDim.x * blockDim.x;
  for (; i < n; i += stride) {
    int c = (int)(i % 3);
    __hip_atomic_fetch_add(&s[c], ea[i] * inv, __ATOMIC_RELAXED, __HIP_MEMORY_SCOPE_WORKGROUP);
  }
  __syncthreads();
  if (threadIdx.x < 3)
    __hip_atomic_fetch_add(&mean_out[threadIdx.x], s[threadIdx.x], __ATOMIC_RELAXED,
                           __HIP_MEMORY_SCOPE_AGENT);
}

// ---------------------------------------------------------------------------
// Pad A [M,K] -> [M,K4] zero-filled (only needed for layer 1: K=2 -> 4)
// ---------------------------------------------------------------------------
__global__ void pad_a_kernel(const float* __restrict__ A, float* __restrict__ Ap,
                             long M, int K, int K4) {
  long i = (long)blockIdx.x * blockDim.x + threadIdx.x;
  long n = M * K4;
  if (i >= n) return;
  long row = i / K4;
  int col = (int)(i - row * K4);
  Ap[i] = (col < K) ? A[row * K + col] : 0.0f;
}

// ---------------------------------------------------------------------------
// Pack W [K,Nt] into WMMA-B fragment order, zero-padded to [K4,NT]:
//   Wp[((t*KS + ks)*32 + lane)*2 + {0,1}] = W[ks*4 + 2*(lane>>4) + {0,1}][t*16 + (lane&15)]
// so the GEMM loads each lane's B pair with a single contiguous b64 load.
// ---------------------------------------------------------------------------
__global__ void pack_w_kernel(const float* __restrict__ W, float* __restrict__ Wp,
                              int K, int K4, int Nt, int NT) {
  int KS = K4 >> 2;
  int total = (NT >> 4) * KS * 32;
  int i = blockIdx.x * blockDim.x + threadIdx.x;
  if (i >= total) return;
  int lane = i & 31;
  int frag = i >> 5;
  int ks = frag % KS;
  int t  = frag / KS;
  int half = lane >> 4, l = lane & 15;
  int n  = t * 16 + l;
  int ka = ks * 4 + 2 * half;
  float w0 = (ka     < K && n < Nt) ? W[(size_t)ka * Nt + n]       : 0.0f;
  float w1 = (ka + 1 < K && n < Nt) ? W[(size_t)(ka + 1) * Nt + n] : 0.0f;
  Wp[(size_t)i * 2 + 0] = w0;
  Wp[(size_t)i * 2 + 1] = w1;
}

__global__ void pad_bias_kernel(const float* __restrict__ b, float* __restrict__ bp,
                                int Nt, int NT) {
  int i = blockIdx.x * blockDim.x + threadIdx.x;
  if (i >= NT) return;
  bp[i] = (i < Nt) ? b[i] : 0.0f;
}

// ---------------------------------------------------------------------------
// C[M,NT] = A[M,K4] @ Wpacked + biasp       (fp32, V_WMMA_F32_16X16X4_F32)
// Compile-time K4/NT: branch-free inner loop, one wave per 16-row strip,
// A fragment reused across all NT/16 column tiles.
// ---------------------------------------------------------------------------
template <int K4, int NT>
__global__ void gemm_wmma_kernel(const float* __restrict__ A, const float* __restrict__ Wp,
                                 const float* __restrict__ biasp, float* __restrict__ C,
                                 int M) {
  constexpr int TN = NT / 16;   // column tiles
  constexpr int KS = K4 / 4;    // k steps
  int wave = blockIdx.x * ((int)blockDim.x >> 5) + ((int)threadIdx.x >> 5);
  if (wave * 16 >= M) return;   // wave-uniform: EXEC stays all-ones for WMMA
  int lane = (int)threadIdx.x & 31;
  int half = lane >> 4;         // selects K pair (k,k+1) vs (k+2,k+3)
  int l    = lane & 15;
  int m = wave * 16 + l;

  v8f zero = {};
  v8f acc[TN];
#pragma unroll
  for (int t = 0; t < TN; ++t) acc[t] = zero;

  const float* Arow = A + (size_t)m * K4 + 2 * half;
#pragma unroll 4
  for (int ks = 0; ks < KS; ++ks) {
    v2f a = *(const v2f*)(Arow + ks * 4);
#pragma unroll
    for (int t = 0; t < TN; ++t) {
      v2f b = *(const v2f*)(Wp + ((size_t)(t * KS + ks) * 32 + lane) * 2);
      acc[t] = __builtin_amdgcn_wmma_f32_16x16x4_f32(false, a, false, b, (short)0,
                                                     acc[t], false, false);
    }
  }
#pragma unroll
  for (int t = 0; t < TN; ++t) {
    int n = t * 16 + l;
    float bv = biasp[n];
#pragma unroll
    for (int r = 0; r < 8; ++r) {
      int mm = wave * 16 + r + 8 * half;
      C[(size_t)mm * NT + n] = acc[t][r] + bv;
    }
  }
}

// ---------------------------------------------------------------------------
// Attention score per (edge, head): e = sum_c leaky(xl[src]+xr[dst]+ea@We)*att
// Running segment max over dst via native u32 atomic max (ordered encoding).
// ldx = row stride of xl/xr (may exceed H*C when output columns are padded).
// ---------------------------------------------------------------------------
__global__ void edge_score_kernel(const float* __restrict__ xl, const float* __restrict__ xr,
                                  const int* __restrict__ ei, const float* __restrict__ ea,
                                  const float* __restrict__ mean_ea,
                                  const float* __restrict__ We, const float* __restrict__ att,
                                  float* __restrict__ ebuf, unsigned* __restrict__ emax,
                                  int E, int Nn, int H, int C, int ldx) {
  long gid = (long)blockIdx.x * blockDim.x + threadIdx.x;
  long total = (long)(E + Nn) * H;
  if (gid >= total) return;
  int eid = (int)(gid / H);
  int h   = (int)(gid - (long)eid * H);
  int src, dst;
  float e0, e1, e2;
  if (eid < E) {
    src = ei[eid];
    dst = ei[E + eid];
    const float* eap = ea + (size_t)eid * 3;
    e0 = eap[0]; e1 = eap[1]; e2 = eap[2];
  } else {               // self-loop, edge_attr filled with column mean
    src = dst = eid - E;
    e0 = mean_ea[0]; e1 = mean_ea[1]; e2 = mean_ea[2];
  }
  int HC = H * C;
  const float* xls = xl + (size_t)src * ldx + h * C;
  const float* xrd = xr + (size_t)dst * ldx + h * C;
  __builtin_prefetch(xls, 0, 0);
  const float* We0 = We + h * C;
  const float* We1 = We0 + HC;
  const float* We2 = We1 + HC;
  const float* at  = att + h * C;
  float s = 0.0f;
  for (int c = 0; c < C; ++c) {
    float ef = e0 * We0[c] + e1 * We1[c] + e2 * We2[c];
    float mv = xls[c] + xrd[c] + ef;
    mv = mv > 0.0f ? mv : 0.2f * mv;   // leaky_relu(0.2)
    s += mv * at[c];
  }
  ebuf[gid] = s;
  __hip_atomic_fetch_max(&emax[(size_t)dst * H + h], encodeOrdered(s),
                         __ATOMIC_RELAXED, __HIP_MEMORY_SCOPE_AGENT);
}

// ---------------------------------------------------------------------------
// ee = exp(e - max[dst]); den[dst] += ee   (native global_atomic_add_f32)
// ---------------------------------------------------------------------------
__global__ void edge_exp_kernel(float* __restrict__ ebuf, const unsigned* __restrict__ emax,
                                float* __restrict__ den, const int* __restrict__ ei,
                                int E, int Nn, int H) {
  long gid = (long)blockIdx.x * blockDim.x + threadIdx.x;
  long total = (long)(E + Nn) * H;
  if (gid >= total) return;
  int eid = (int)(gid / H);
  int h   = (int)(gid - (long)eid * H);
  int dst = (eid < E) ? ei[E + eid] : (eid - E);
  float mx = decodeOrdered(emax[(size_t)dst * H + h]);
  float v = __expf(ebuf[gid] - mx);
  ebuf[gid] = v;
  __hip_atomic_fetch_add(&den[(size_t)dst * H + h], v,
                         __ATOMIC_RELAXED, __HIP_MEMORY_SCOPE_AGENT);
}

// ---------------------------------------------------------------------------
// out[dst] += (ee/den[dst]) * xl[src]   per channel, scatter with f32 atomics
// out has row stride H*C (unpadded).
// ---------------------------------------------------------------------------
__global__ void edge_agg_kernel(const float* __restrict__ ebuf, const float* __restrict__ den,
                                const float* __restrict__ xl, const int* __restrict__ ei,
                                float* __restrict__ out, int E, int Nn, int H, int C, int ldx) {
  long gid = (long)blockIdx.x * blockDim.x + threadIdx.x;
  long total = (long)(E + Nn) * H;
  if (gid >= total) return;
  int eid = (int)(gid / H);
  int h   = (int)(gid - (long)eid * H);
  int src, dst;
  if (eid < E) { src = ei[eid]; dst = ei[E + eid]; }
  else         { src = dst = eid - E; }
  int HC = H * C;
  float alpha = ebuf[gid] / den[(size_t)dst * H + h];
  const float* xls = xl + (size_t)src * ldx + h * C;
  float* o = out + (size_t)dst * HC + h * C;
  for (int c = 0; c < C; ++c)
    __hip_atomic_fetch_add(&o[c], alpha * xls[c],
                           __ATOMIC_RELAXED, __HIP_MEMORY_SCOPE_AGENT);
}

// ---------------------------------------------------------------------------
// out = acc + bias (per channel), optional ELU
// ---------------------------------------------------------------------------
__global__ void bias_act_kernel(const float* __restrict__ acc, const float* __restrict__ bias,
                                float* __restrict__ out, long n, int HC, int do_elu) {
  long i = (long)blockIdx.x * blockDim.x + threadIdx.x;
  if (i >= n) return;
  int c = (int)(i % HC);
  float v = acc[i] + bias[c];
  if (do_elu) v = (v > 0.0f) ? v : (__expf(v) - 1.0f);
  out[i] = v;
}

// ---------------------------------------------------------------------------
// Host-side orchestration
// ---------------------------------------------------------------------------
extern "C" void kernel_launch(void* const* d_in, const int* in_sizes, int n_in,
                              void* d_out, int out_size, void* d_ws, size_t ws_size,
                              hipStream_t stream) {
  const float* x  = (const float*)d_in[0];
  const int*   ei = (const int*)d_in[1];
  const float* ea = (const float*)d_in[2];
  const float* P[21];
  for (int i = 0; i < 21; ++i) P[i] = (const float*)d_in[3 + i];
  const float *Wl1 = P[0],  *bl1 = P[1],  *Wr1 = P[2],  *br1 = P[3],  *We1 = P[4],  *att1 = P[5],  *b1 = P[6];
  const float *Wl2 = P[7],  *bl2 = P[8],  *Wr2 = P[9],  *br2 = P[10], *We2 = P[11], *att2 = P[12], *b2 = P[13];
  const float *Wl3 = P[14], *bl3 = P[15], *Wr3 = P[16], *br3 = P[17], *We3 = P[18], *att3 = P[19], *b3 = P[20];

  const int N    = in_sizes[0] / 2;
  const int E    = in_sizes[2] / 3;
  const int Etot = E + N;

  float* ws = (float*)d_ws;
  size_t o = 0;
  float*    B0     = ws + o; o += (size_t)N * 128;   // xl
  float*    B1     = ws + o; o += (size_t)N * 128;   // xr
  float*    B2     = ws + o; o += (size_t)N * 128;   // aggregation acc / activations
  float*    Ebuf   = ws + o; o += (size_t)Etot * 8;  // per-(edge,head) scores
  unsigned* Emax   = (unsigned*)(ws + o); o += (size_t)N * 8;
  float*    Den    = ws + o; o += (size_t)N * 8;
  float*    Xpad   = ws + o; o += (size_t)N * 4;     // layer-1 A padded K=2->4
  float*    WpL    = ws + o; o += 16384;             // packed Wl (max 8192 used)
  float*    WpR    = ws + o; o += 16384;             // packed Wr
  float*    BpL    = ws + o; o += 128;               // padded bl
  float*    BpR    = ws + o; o += 128;               // padded br
  float*    MeanEA = ws + o; o += 4;
  (void)ws_size; (void)n_in; (void)out_size;

  const int TB = 256;
  auto zero = [&](float* p, long n) {
    int blocks = (int)((n + TB - 1) / TB);
    if (blocks > 16384) blocks = 16384;
    zero_kernel<<<blocks, TB, 0, stream>>>(p, n);
  };
  auto packW = [&](const float* W, const float* b, float* Wp, float* bp,
                   int K, int K4, int Nt, int NT) {
    int total = (NT >> 4) * (K4 >> 2) * 32;
    pack_w_kernel<<<(total + TB - 1) / TB, TB, 0, stream>>>(W, Wp, K, K4, Nt, NT);
    pad_bias_kernel<<<1, 128, 0, stream>>>(b, bp, Nt, NT);
  };

  // edge_attr column mean for self-loop fill
  zero(MeanEA, 4);
  {
    long n = (long)E * 3;
    int blocks = (int)((n + TB - 1) / TB);
    if (blocks > 16384) blocks = 16384;
    mean_ea_kernel<<<blocks, TB, 0, stream>>>(ea, MeanEA, E);
  }
  // pad layer-1 features [N,2] -> [N,4]
  {
    long n = (long)N * 4;
    pad_a_kernel<<<(int)((n + TB - 1) / TB), TB, 0, stream>>>(x, Xpad, N, 2, 4);
  }

  int gemm_blocks = (int)(((long)((N + 15) / 16) + 7) / 8);  // 8 waves/block

  auto run_edges = [&](int H, int C, int ldx, const float* We, const float* att,
                       const float* b, int do_elu, float* final_out) {
    const int HC = H * C;
    zero(B2, (long)N * HC);
    zero((float*)Emax, (long)N * H);   // 0u == identity for ordered-encoded max
    zero(Den, (long)N * H);
    long tot = (long)Etot * H;
    int eblocks = (int)((tot + TB - 1) / TB);
    edge_score_kernel<<<eblocks, TB, 0, stream>>>(B0, B1, ei, ea, MeanEA, We, att,
                                                  Ebuf, Emax, E, N, H, C, ldx);
    edge_exp_kernel<<<eblocks, TB, 0, stream>>>(Ebuf, Emax, Den, ei, E, N, H);
    edge_agg_kernel<<<eblocks, TB, 0, stream>>>(Ebuf, Den, B0, ei, B2, E, N, H, C, ldx);
    long nn = (long)N * HC;
    bias_act_kernel<<<(int)((nn + TB - 1) / TB), TB, 0, stream>>>(B2, b, final_out,
                                                                  nn, HC, do_elu);
  };

  // ---- Layer 1: in=2 (K4=4), H=8, C=16 -> 128, ELU --------------------------
  packW(Wl1, bl1, WpL, BpL, 2, 4, 128, 128);
  packW(Wr1, br1, WpR, BpR, 2, 4, 128, 128);
  gemm_wmma_kernel<4, 128><<<gemm_blocks, 256, 0, stream>>>(Xpad, WpL, BpL, B0, N);
  gemm_wmma_kernel<4, 128><<<gemm_blocks, 256, 0, stream>>>(Xpad, WpR, BpR, B1, N);
  run_edges(8, 16, 128, We1, att1, b1, 1, B2);          // h1 in B2 [N,128]

  // ---- Layer 2: in=128, H=4, C=16 -> 64, ELU --------------------------------
  packW(Wl2, bl2, WpL, BpL, 128, 128, 64, 64);
  packW(Wr2, br2, WpR, BpR, 128, 128, 64, 64);
  gemm_wmma_kernel<128, 64><<<gemm_blocks, 256, 0, stream>>>(B2, WpL, BpL, B0, N);
  gemm_wmma_kernel<128, 64><<<gemm_blocks, 256, 0, stream>>>(B2, WpR, BpR, B1, N);
  run_edges(4, 16, 64, We2, att2, b2, 1, B2);           // h2 in B2 [N,64]

  // ---- Layer 3: in=64, H=1, C=2 -> 2 (cols padded to 16), no ELU ------------
  packW(Wl3, bl3, WpL, BpL, 64, 64, 2, 16);
  packW(Wr3, br3, WpR, BpR, 64, 64, 2, 16);
  gemm_wmma_kernel<64, 16><<<gemm_blocks, 256, 0, stream>>>(B2, WpL, BpL, B0, N);
  gemm_wmma_kernel<64, 16><<<gemm_blocks, 256, 0, stream>>>(B2, WpR, BpR, B1, N);
  run_edges(1, 2, 16, We3, att3, b3, 0, (float*)d_out); // out [N,2]
}